// LightGCN_33371895890154
// MI455X (gfx1250) — compile-verified
//
#include <hip/hip_runtime.h>
#include <hip/hip_bf16.h>

// LightGCN on MI455X (gfx1250).
// Bandwidth/atomic-bound SpMM (COO scatter-add), ~0.4 FLOP/byte -> WMMA does
// not apply (no dense matmul structure; 16x16 tiling would require per-call
// edge sorting that costs more than the atomic SpMM). CDNA5 features used:
// async global->LDS B128 staging of edge indices + precomputed edge norms
// (ASYNCcnt + s_wait_asynccnt), LDS broadcast reads, native
// global_atomic_add_f32 scatter, global_prefetch_b8 for upcoming gather rows.
//
// Roofline: per layer the gather touches 614 MB of L2 traffic but only 38.4 MB
// of distinct data (fits in 192 MB L2); HBM-compulsory traffic is ~200 MB total
// -> runtime is bounded by L2 atomic/gather throughput, not HBM.

#define TPB   256
#define CHUNK 1024                 // edges staged per block (4 per thread via B128)
#define EPW   (CHUNK / (TPB / 32)) // 128 edges per wave

typedef __attribute__((address_space(3))) int         lds_i32;  // 32-bit LDS ptr
typedef __attribute__((address_space(3))) float       lds_f32;
typedef __attribute__((address_space(1))) const int   g_ci32;   // 64-bit global ptr
typedef __attribute__((address_space(1))) const float g_cf32;

__global__ __launch_bounds__(TPB) void lgcn_zero_deg(unsigned* deg, int n) {
    int i = blockIdx.x * blockDim.x + threadIdx.x;
    if (i < n) deg[i] = 0u;
}

__global__ __launch_bounds__(TPB) void lgcn_count_deg(const int* __restrict__ to,
                                                      unsigned* __restrict__ deg, int E) {
    int e = blockIdx.x * blockDim.x + threadIdx.x;
    if (e < E) atomicAdd(&deg[to[e]], 1u);
}

__global__ __launch_bounds__(TPB) void lgcn_rsqrt_deg(float* dinv, int n) {
    int i = blockIdx.x * blockDim.x + threadIdx.x;
    if (i < n) {
        unsigned c = ((const unsigned*)dinv)[i];
        dinv[i] = (c > 0u) ? (1.0f / sqrtf((float)c)) : 0.0f;
    }
}

// norm[e] = dinv[frm[e]] * dinv[to[e]]  (layer-invariant; computed once)
__global__ __launch_bounds__(TPB) void lgcn_norm(const int* __restrict__ frm,
                                                 const int* __restrict__ to,
                                                 const float* __restrict__ dinv,
                                                 float* __restrict__ nrm, int E) {
    int e = blockIdx.x * blockDim.x + threadIdx.x;
    if (e < E) nrm[e] = dinv[frm[e]] * dinv[to[e]];
}

// out0 = acc = emb ; zeroA = 0   (vectorized: 16 float4 per node row)
__global__ __launch_bounds__(TPB) void lgcn_init(const float4* __restrict__ emb,
                                                 float4* __restrict__ out0,
                                                 float4* __restrict__ acc,
                                                 float4* __restrict__ zeroA, int n4) {
    int i = blockIdx.x * blockDim.x + threadIdx.x;
    if (i < n4) {
        float4 v = emb[i];
        out0[i] = v;
        acc[i]  = v;
        zeroA[i] = make_float4(0.f, 0.f, 0.f, 0.f);
    }
}

// acc = (acc + lay) * scale ; optionally zero the next layer's destination buffer
__global__ __launch_bounds__(TPB) void lgcn_accum(float4* __restrict__ acc,
                                                  const float4* __restrict__ lay,
                                                  float4* zbuf, float scale, int n4) {
    int i = blockIdx.x * blockDim.x + threadIdx.x;
    if (i < n4) {
        float4 a = acc[i];
        float4 b = lay[i];
        a.x = (a.x + b.x) * scale;
        a.y = (a.y + b.y) * scale;
        a.z = (a.z + b.z) * scale;
        a.w = (a.w + b.w) * scale;
        acc[i] = a;
        if (zbuf) zbuf[i] = make_float4(0.f, 0.f, 0.f, 0.f);
    }
}

// one edge: all 32 lanes cooperate; float2 of the 64-wide row per lane
__device__ __forceinline__ void lgcn_edge(int f, int t, float nrm,
                                          const float* __restrict__ x,
                                          float* __restrict__ y, int lo) {
    const float2 v = *(const float2*)(x + (size_t)f * 64 + lo); // global_load_b64
    float* dst = y + (size_t)t * 64 + lo;
    unsafeAtomicAdd(dst,     v.x * nrm); // global_atomic_add_f32 (no return)
    unsafeAtomicAdd(dst + 1, v.y * nrm);
}

// y[to] += norm * x[frm]
__global__ __launch_bounds__(TPB) void lgcn_scatter(const int* __restrict__ frm,
                                                    const int* __restrict__ to,
                                                    const float* __restrict__ nrm,
                                                    const float* __restrict__ x,
                                                    float* __restrict__ y, int E) {
    __shared__ int   frm_s[CHUNK];
    __shared__ int   to_s[CHUNK];
    __shared__ float nrm_s[CHUNK];

    const int tid  = threadIdx.x;
    const int base = blockIdx.x * CHUNK;

    // --- async DMA stage: 4 frm + 4 to + 4 norm per thread (B128 each) ---
    // (E is a multiple of 4; norm buffer is padded, frm overreads land in `to`)
    {
        int e = base + tid * 4;
        if (e < E) {
            lds_i32* lf = (lds_i32*)&frm_s[tid * 4];
            lds_i32* lt = (lds_i32*)&to_s[tid * 4];
            lds_f32* ln = (lds_f32*)&nrm_s[tid * 4];
            g_ci32*  gf = (g_ci32*)(frm + e);
            g_ci32*  gt = (g_ci32*)(to + e);
            g_cf32*  gn = (g_cf32*)(nrm + e);
            asm volatile("global_load_async_to_lds_b128 %0, %1, off\n\t"
                         "global_load_async_to_lds_b128 %2, %3, off\n\t"
                         "global_load_async_to_lds_b128 %4, %5, off"
                         :
                         : "v"(lf), "v"(gf), "v"(lt), "v"(gt), "v"(ln), "v"(gn)
                         : "memory");
        }
        // each wave consumes only edges staged by its own lanes -> no barrier
        asm volatile("s_wait_asynccnt 0x0" ::: "memory");
    }

    const int wave = tid >> 5;
    const int lane = tid & 31;
    const int ebeg = wave * EPW;
    const int lo   = lane * 2; // 2 floats per lane covers D=64

    if (base + CHUNK <= E) {
        // steady state: full chunk, branch-free, unrolled for MLP
#pragma unroll 2
        for (int i = 0; i < EPW; ++i) {
            // prefetch an upcoming gather row (covers this lane's cacheline)
            int ip = i + 8;
            if (ip < EPW)
                __builtin_prefetch(x + (size_t)frm_s[ebeg + ip] * 64 + lo, 0, 0);
            lgcn_edge(frm_s[ebeg + i], to_s[ebeg + i], nrm_s[ebeg + i], x, y, lo);
        }
    } else {
        for (int i = 0; i < EPW; ++i) {
            if (base + ebeg + i >= E) break;
            lgcn_edge(frm_s[ebeg + i], to_s[ebeg + i], nrm_s[ebeg + i], x, y, lo);
        }
    }
}

extern "C" void kernel_launch(void* const* d_in, const int* in_sizes, int n_in,
                              void* d_out, int out_size, void* d_ws, size_t ws_size,
                              hipStream_t stream) {
    const int    E   = in_sizes[0] / 2;       // edge_list is [2, E]
    const int*   frm = (const int*)d_in[0];
    const int*   to  = frm + E;
    const int    N   = in_sizes[1] / 64;      // embedding is [N, 64]
    const int    n4  = N * 16;                // float4 elements per pass
    const float* emb = (const float*)d_in[1];

    // workspace: [deg/dinv: N f32][norm: E f32][A: N*64][B: N*64]  (~87 MB)
    char*  ws        = (char*)d_ws;
    size_t degBytes  = ((size_t)N * 4 + 4095) & ~(size_t)4095;
    size_t normBytes = ((size_t)E * 4 + 4095) & ~(size_t)4095;
    float* dinv      = (float*)ws;
    float* nrm       = (float*)(ws + degBytes);
    float* A         = (float*)(ws + degBytes + normBytes);
    float* B         = A + (size_t)N * 64;

    float* out0 = (float*)d_out;              // emb0 copy
    float* acc  = out0 + (size_t)N * 64;      // running sum -> mean

    dim3 blk(TPB);
    int gN = (N + TPB - 1) / TPB;
    int gE = (E + TPB - 1) / TPB;
    int g4 = (n4 + TPB - 1) / TPB;
    int gS = (E + CHUNK - 1) / CHUNK;

    // degree -> deg^-1/2 (in place: u32 counts reinterpreted, then float),
    // then per-edge norms (layer-invariant)
    lgcn_zero_deg<<<gN, blk, 0, stream>>>((unsigned*)dinv, N);
    lgcn_count_deg<<<gE, blk, 0, stream>>>(to, (unsigned*)dinv, E);
    lgcn_rsqrt_deg<<<gN, blk, 0, stream>>>(dinv, N);
    lgcn_norm<<<gE, blk, 0, stream>>>(frm, to, dinv, nrm, E);

    // out0 = acc = emb0 ; A = 0
    lgcn_init<<<g4, blk, 0, stream>>>((const float4*)emb, (float4*)out0,
                                      (float4*)acc, (float4*)A, n4);

    // layer 1: emb0 -> A ; acc += A ; B = 0
    lgcn_scatter<<<gS, blk, 0, stream>>>(frm, to, nrm, emb, A, E);
    lgcn_accum<<<g4, blk, 0, stream>>>((float4*)acc, (const float4*)A, (float4*)B, 1.0f, n4);

    // layer 2: A -> B ; acc += B ; A = 0
    lgcn_scatter<<<gS, blk, 0, stream>>>(frm, to, nrm, A, B, E);
    lgcn_accum<<<g4, blk, 0, stream>>>((float4*)acc, (const float4*)B, (float4*)A, 1.0f, n4);

    // layer 3: B -> A ; acc = (acc + A) * 0.25  (mean over 4 stages)
    lgcn_scatter<<<gS, blk, 0, stream>>>(frm, to, nrm, B, A, E);
    lgcn_accum<<<g4, blk, 0, stream>>>((float4*)acc, (const float4*)A, nullptr, 0.25f, n4);
}